// BMEDModel_9113920602606
// MI455X (gfx1250) — compile-verified
//
#include <hip/hip_runtime.h>
#include <hip/hip_bf16.h>

#define DTSTEP 0.1f
#define HDIM 192
#define NW 12                 // 12 waves: one 16-column WMMA tile each
#define NTHREADS (NW * 32)
#define NCHUNK 48             // 192 / 4 K-chunks per 192->192 layer

typedef __attribute__((ext_vector_type(2))) float v2f;
typedef __attribute__((ext_vector_type(8))) float v8f;

// Only probe for the builtin in the device pass; the host (x86) pass does not
// register amdgcn builtins for __has_builtin even though it accepts them in
// __device__ bodies via the aux-target mechanism.
#if defined(__HIP_DEVICE_COMPILE__)
#if !__has_builtin(__builtin_amdgcn_wmma_f32_16x16x4_f32)
#error "device pass: missing __builtin_amdgcn_wmma_f32_16x16x4_f32"
#endif
#endif

static __device__ __forceinline__ v8f wmma4(v2f a, v2f b, v8f c) {
  // D = A(16x4 f32) * B(4x16 f32) + C(16x16 f32)
  return __builtin_amdgcn_wmma_f32_16x16x4_f32(
      /*neg_a=*/false, a, /*neg_b=*/false, b,
      /*c_mod=*/(short)0, c, /*reuse_a=*/false, /*reuse_b=*/false);
}

__global__ __launch_bounds__(NTHREADS, 1)
void bmed_scan_kernel(const float* __restrict__ init_state,
                      const float* __restrict__ times, int n_times,
                      const float* __restrict__ meanv,
                      const float* __restrict__ scalev,
                      const float* __restrict__ W1, const float* __restrict__ b1,
                      const float* __restrict__ W2, const float* __restrict__ b2,
                      const float* __restrict__ W3, const float* __restrict__ b3,
                      const float* __restrict__ W4, const float* __restrict__ b4,
                      float* __restrict__ out)
{
  __shared__ __align__(16) float hA[HDIM];
  __shared__ __align__(16) float hB[HDIM];
  __shared__ float xn[8];
  __shared__ float st[16];
  __shared__ float mig[4];
  __shared__ float W1L[7 * HDIM];
  __shared__ float b1L[HDIM];
  __shared__ float W4L[HDIM * 4];
  __shared__ float b4L[4];
  __shared__ float meanL[8];
  __shared__ float scaleL[8];

  const int tid  = threadIdx.x;
  const int wv   = tid >> 5;
  const int lane = tid & 31;
  const int col  = wv * 16 + (lane & 15);     // output column owned by this lane's half
  const int roff = (lane < 16) ? 0 : 2;       // K sub-offset per f32 A/B VGPR layout

  // ---------------- one-time setup ----------------
  for (int i = tid; i < 7 * HDIM; i += NTHREADS) W1L[i] = W1[i];
  for (int i = tid; i < HDIM;     i += NTHREADS) b1L[i] = b1[i];
  for (int i = tid; i < HDIM * 4; i += NTHREADS) W4L[i] = W4[i];
  if (tid < 4)  b4L[tid] = b4[tid];
  if (tid < 7)  { meanL[tid] = meanv[tid]; scaleL[tid] = scalev[tid]; }
  if (tid < 10) { st[tid] = init_state[tid]; out[tid] = init_state[tid]; }

  // Register-resident B operands for this wave's 16-column tile of W2/W3.
  // Chunk c covers rows 4c..4c+3: lanes 0-15 hold rows {4c,4c+1},
  // lanes 16-31 hold rows {4c+2,4c+3} (mirrors the documented f32 A layout).
  v2f w2b[NCHUNK], w3b[NCHUNK];
#pragma unroll
  for (int c = 0; c < NCHUNK; ++c) {
    const int r = 4 * c + roff;
    v2f t2; t2.x = W2[r * HDIM + col]; t2.y = W2[(r + 1) * HDIM + col];
    v2f t3; t3.x = W3[r * HDIM + col]; t3.y = W3[(r + 1) * HDIM + col];
    w2b[c] = t2;
    w3b[c] = t3;
  }
  const float bias2 = b2[col];
  const float bias3 = b3[col];

  const int n_steps = __float2int_rn(times[n_times - 1] / DTSTEP);

  // measured_indices: [0, round(t/dt)...] as int32 after the float block
  {
    int* mi = (int*)(out + (size_t)(n_steps + 1) * 10);
    if (tid == 0) mi[0] = 0;
    if (tid < n_times) mi[1 + tid] = __float2int_rn(times[tid] / DTSTEP);
  }

  __syncthreads();

  // ---------------- sequential scan ----------------
  for (int step = 0; step < n_steps; ++step) {
    // normalized observation (dropout identity in eval; stop_gradient is a no-op fwd)
    if (tid < 7) xn[tid] = (st[tid] - meanL[tid]) / scaleL[tid];
    __syncthreads();

    // Layer 1: hA = relu(x @ W1 + b1)   (K=7, pure VALU)
    if (tid < HDIM) {
      float s = b1L[tid];
#pragma unroll
      for (int k = 0; k < 7; ++k) s = fmaf(xn[k], W1L[k * HDIM + tid], s);
      hA[tid] = fmaxf(s, 0.0f);
    }
    __syncthreads();

    // Layer 2: hB = relu(hA @ W2 + b2)  — f32 WMMA, 4 independent accumulators
    {
      v8f a0 = {}, a1 = {}, a2 = {}, a3 = {};
#pragma unroll
      for (int c = 0; c < NCHUNK; c += 4) {
        v2f x0 = *(const v2f*)(hA + 4 * (c + 0) + roff);
        a0 = wmma4(x0, w2b[c + 0], a0);
        v2f x1 = *(const v2f*)(hA + 4 * (c + 1) + roff);
        a1 = wmma4(x1, w2b[c + 1], a1);
        v2f x2 = *(const v2f*)(hA + 4 * (c + 2) + roff);
        a2 = wmma4(x2, w2b[c + 2], a2);
        v2f x3 = *(const v2f*)(hA + 4 * (c + 3) + roff);
        a3 = wmma4(x3, w2b[c + 3], a3);
      }
      // D row M=0 lives in VGPR0 (element 0), lanes 0-15
      float r = ((a0[0] + a1[0]) + (a2[0] + a3[0])) + bias2;
      r = fmaxf(r, 0.0f);
      if (lane < 16) hB[col] = r;
    }
    __syncthreads();

    // Layer 3: hA = relu(hB @ W3 + b3)
    {
      v8f a0 = {}, a1 = {}, a2 = {}, a3 = {};
#pragma unroll
      for (int c = 0; c < NCHUNK; c += 4) {
        v2f x0 = *(const v2f*)(hB + 4 * (c + 0) + roff);
        a0 = wmma4(x0, w3b[c + 0], a0);
        v2f x1 = *(const v2f*)(hB + 4 * (c + 1) + roff);
        a1 = wmma4(x1, w3b[c + 1], a1);
        v2f x2 = *(const v2f*)(hB + 4 * (c + 2) + roff);
        a2 = wmma4(x2, w3b[c + 2], a2);
        v2f x3 = *(const v2f*)(hB + 4 * (c + 3) + roff);
        a3 = wmma4(x3, w3b[c + 3], a3);
      }
      float r = ((a0[0] + a1[0]) + (a2[0] + a3[0])) + bias3;
      r = fmaxf(r, 0.0f);
      if (lane < 16) hA[col] = r;
    }
    __syncthreads();

    // Layer 4: mig = hA @ W4 + b4   (N=4, pure VALU)
    if (tid < 4) {
      float s = b4L[tid];
#pragma unroll 8
      for (int k = 0; k < HDIM; ++k) s = fmaf(hA[k], W4L[k * 4 + tid], s);
      mig[tid] = s;
    }
    __syncthreads();

    // state update + stream trajectory row
    if (tid == 0) {
      const float dNLA = mig[0] * DTSTEP, dNK = mig[1] * DTSTEP;
      const float dVA  = mig[2] * DTSTEP, dVB = mig[3] * DTSTEP;
      const float VF = st[7], VA = st[8], VB = st[9];
      const float nVF = VF - dVA - dVB;
      const float nVA = VA + dVA;
      const float nVB = VB + dVB;
      const float c3 = (st[3] * VF - dNLA) / nVF;
      const float c4 = (st[4] * VA + dNLA) / nVA;
      const float c5 = (st[5] * VF - dNK) / nVF;
      const float c6 = (st[6] * VB + dNK) / nVB;
      st[3] = c3; st[4] = c4; st[5] = c5; st[6] = c6;
      st[7] = nVF; st[8] = nVA; st[9] = nVB;
      float* row = out + (size_t)(step + 1) * 10;
      row[0] = st[0]; row[1] = st[1]; row[2] = st[2];
      row[3] = c3; row[4] = c4; row[5] = c5; row[6] = c6;
      row[7] = nVF; row[8] = nVA; row[9] = nVB;
    }
    __syncthreads();
  }
}

extern "C" void kernel_launch(void* const* d_in, const int* in_sizes, int n_in,
                              void* d_out, int out_size, void* d_ws, size_t ws_size,
                              hipStream_t stream) {
  (void)n_in; (void)out_size; (void)d_ws; (void)ws_size;
  const float* init_state = (const float*)d_in[0];
  const float* times      = (const float*)d_in[1];
  const float* meanv      = (const float*)d_in[2];
  const float* scalev     = (const float*)d_in[3];
  const float* W1 = (const float*)d_in[4];
  const float* b1 = (const float*)d_in[5];
  const float* W2 = (const float*)d_in[6];
  const float* b2 = (const float*)d_in[7];
  const float* W3 = (const float*)d_in[8];
  const float* b3 = (const float*)d_in[9];
  const float* W4 = (const float*)d_in[10];
  const float* b4 = (const float*)d_in[11];

  // One persistent workgroup: the scan is a hard serial dependency chain,
  // so one WGP with register/LDS-resident weights is the right shape.
  bmed_scan_kernel<<<1, NTHREADS, 0, stream>>>(
      init_state, times, in_sizes[1], meanv, scalev,
      W1, b1, W2, b2, W3, b3, W4, b4, (float*)d_out);
}